// LatentSDE_36361193128532
// MI455X (gfx1250) — compile-verified
//
#include <hip/hip_runtime.h>

#define E_ 8
#define S_ 6000
#define D_ 128
#define A_ 16
#define L_ 64
#define H_ 256
#define C_ 64
#define B_ 50
#define T_ 120
#define NSUB_ 2
#define DT_ 0.5f
#define SQDT_ 0.70710678118654752f

typedef __attribute__((ext_vector_type(2))) float v2f;
typedef __attribute__((ext_vector_type(4))) float v4f;
typedef __attribute__((ext_vector_type(8))) float v8f;

// ---- LDS layout (floats). Total 16384 floats = 64 KB, heavy aliasing. ----
#define OFF_SDW 0        // 1024  (sde dW*sqdt)        [aliases SX]
#define OFF_SHZ 1024     // 1024  (drift h output)     [aliases SX]
#define OFF_SX  0        // 2048  (x tile, dead before SDW/SHZ used)
#define OFF_SH  2048     // 4096  (16x256 buf; also lat 16x208)
#define OFF_SH2 6144     // 4096  (16x256 buf; also q 16x128 at +0, ctx 16x64 at +2048)
#define OFF_SZC 10240    // 1024
#define OFF_SZH 11264    // 1024
#define OFF_FZ0 12288    // 1024
#define OFF_FZ1 13312    // 1024
#define OFF_GZ0 14336    // 1024
#define OFF_GZ1 15360    // 1024
#define SMEM_FLOATS 16384

struct Params {
  const float *xs, *actions, *z0_noise, *dW;
  const float *enc_w1, *enc_b1, *enc_w2, *enc_b2, *qz0_w, *qz0_b, *act_w, *act_b;
  const float *fw1, *fb1, *fw2, *fb2, *fw3, *fb3;
  const float *hw1, *hb1, *hw2, *hb2, *hw3, *hb3;
  const float *gw1, *gb1, *gw2, *gb2;
  const float *pw1, *pb1, *pw2, *pb2, *pw3, *pb3;
  const float *pz0_mean, *pz0_logstd;
  float *out;      // [E] logqp, then [E,S,D] predicted_xs
  float *zstate;   // [E*B*L] scratch in ws
  float *acc;      // [E*B]   scratch in ws
  int t;
};

__device__ __forceinline__ float sigmoidf(float x) { return 1.0f / (1.0f + __expf(-x)); }

// Wave-level GEMM: C(16x16) = A(16xK, row-major LDS, stride lda) x B(KxN, row-major global,
// stride ldb), output columns [ncol, ncol+16). Uses V_WMMA_F32_16X16X4_F32.
// A frag (ISA 7.12.2, 32-bit A 16x4): lane<16 -> K={k0,k0+1}; lane>=16 -> K={k0+2,k0+3}; M=lane&15.
// B frag (4x16, rows striped across lanes per VGPR): mirrored K pairing, N=lane&15.
__device__ __forceinline__ v8f wmma_gemm_tile(const float* A, int lda,
                                              const float* __restrict__ Bm, int ldb,
                                              int ncol, int K) {
  const int lane = threadIdx.x & 31;
  const int row  = lane & 15;
  const int kh   = (lane >> 4) << 1;     // 0 or 2
  const int col  = ncol + (lane & 15);
  v8f acc = {0.f, 0.f, 0.f, 0.f, 0.f, 0.f, 0.f, 0.f};
  for (int k0 = 0; k0 < K; k0 += 4) {
    v2f a, b;
    a.x = A[row * lda + k0 + kh];
    a.y = A[row * lda + k0 + kh + 1];
    b.x = Bm[(size_t)(k0 + kh) * ldb + col];
    b.y = Bm[(size_t)(k0 + kh + 1) * ldb + col];
    acc = __builtin_amdgcn_wmma_f32_16x16x4_f32(false, a, false, b, (short)0, acc,
                                                false, false);
  }
  return acc;
}

// C/D layout: acc[i] holds M = i + (lane>=16 ? 8 : 0), N = ncol + (lane&15).
__device__ __forceinline__ void store_tile_lds(float* Dst, int ldd, int ncol, v8f acc,
                                               const float* __restrict__ bias, bool sig) {
  const int lane = threadIdx.x & 31;
  const int mb   = (lane >> 4) << 3;
  const int col  = ncol + (lane & 15);
  const float bv = bias[col];
#pragma unroll
  for (int i = 0; i < 8; ++i) {
    float v = acc[i] + bv;
    if (sig) v = sigmoidf(v);
    Dst[(mb + i) * ldd + col] = v;
  }
}

// drift_all: from z (16xL in LDS) compute fz (16xL), gz (16xL); return lf (valid for tid<16).
__device__ float drift_eval(const Params& p, float* sm, const float* z,
                            float* fz, float* gz) {
  float* sh  = sm + OFF_SH;
  float* sh2 = sm + OFF_SH2;
  float* shz = sm + OFF_SHZ;
  const int tid  = threadIdx.x;
  const int wave = tid >> 5;

  // f MLP: L->H (sig) -> H (sig) -> L
  { v8f a = wmma_gemm_tile(z, L_, p.fw1, H_, wave * 16, L_);
    store_tile_lds(sh, H_, wave * 16, a, p.fb1, true); }
  __syncthreads();
  { v8f a = wmma_gemm_tile(sh, H_, p.fw2, H_, wave * 16, H_);
    store_tile_lds(sh2, H_, wave * 16, a, p.fb2, true); }
  __syncthreads();
  if (wave < 4) {
    v8f a = wmma_gemm_tile(sh2, H_, p.fw3, L_, wave * 16, H_);
    store_tile_lds(fz, L_, wave * 16, a, p.fb3, false);
  }
  __syncthreads();
  // h MLP
  { v8f a = wmma_gemm_tile(z, L_, p.hw1, H_, wave * 16, L_);
    store_tile_lds(sh, H_, wave * 16, a, p.hb1, true); }
  __syncthreads();
  { v8f a = wmma_gemm_tile(sh, H_, p.hw2, H_, wave * 16, H_);
    store_tile_lds(sh2, H_, wave * 16, a, p.hb2, true); }
  __syncthreads();
  if (wave < 4) {
    v8f a = wmma_gemm_tile(sh2, H_, p.hw3, L_, wave * 16, H_);
    store_tile_lds(shz, L_, wave * 16, a, p.hb3, false);
  }
  // g network (elementwise over (row,l), reduce over H), float4-vectorized weight streams
  for (int idx = tid; idx < 16 * L_; idx += 512) {
    const int l = idx & (L_ - 1);
    const float zv = z[idx];
    const v4f* __restrict__ g1 = (const v4f*)(p.gw1 + l * H_);
    const v4f* __restrict__ gb = (const v4f*)(p.gb1 + l * H_);
    const v4f* __restrict__ g2 = (const v4f*)(p.gw2 + l * H_);
    float s = 0.f;
    for (int j = 0; j < H_ / 4; ++j) {
      const v4f a = g1[j];
      const v4f b = gb[j];
      const v4f c = g2[j];
      s += c.x * sigmoidf(zv * a.x + b.x);
      s += c.y * sigmoidf(zv * a.y + b.y);
      s += c.z * sigmoidf(zv * a.z + b.z);
      s += c.w * sigmoidf(zv * a.w + b.w);
    }
    gz[idx] = s + p.gb2[l];
  }
  __syncthreads();
  // lf = 0.5 * sum_l ((fz - hz)/gz)^2, one thread per row (deterministic)
  float lf = 0.f;
  if (tid < 16) {
    for (int l = 0; l < L_; ++l) {
      float u = (fz[tid * L_ + l] - shz[tid * L_ + l]) / gz[tid * L_ + l];
      lf += u * u;
    }
    lf *= 0.5f;
  }
  __syncthreads();
  return lf;
}

__global__ __launch_bounds__(512) void sde_step_kernel(Params p) {
  extern __shared__ float sm[];
  const int tid  = threadIdx.x;
  const int wave = tid >> 5;
  const int e    = blockIdx.x >> 2;
  const int r0   = (blockIdx.x & 3) * 16;
  const int t    = p.t;

  float* sx   = sm + OFF_SX;
  float* sh   = sm + OFF_SH;
  float* sh2  = sm + OFF_SH2;
  float* slat = sm + OFF_SH;           // alias (16x208 <= 16x256)
  float* sq   = sm + OFF_SH2;          // alias (16x128)
  float* sctx = sm + OFF_SH2 + 2048;   // alias (16x64)
  float* szc  = sm + OFF_SZC;
  float* szh  = sm + OFF_SZH;
  float* sdw  = sm + OFF_SDW;

  // stage 0: load x tile (zero-padded rows), float4
  for (int i = tid; i < 16 * (D_ / 4); i += 512) {
    const int row = i >> 5, d4 = i & 31;
    const int b = r0 + row;
    v4f v = {0.f, 0.f, 0.f, 0.f};
    if (b < B_) v = *(const v4f*)(p.xs + ((size_t)e * S_ + (size_t)t * B_ + b) * D_ + d4 * 4);
    *(v4f*)(sx + row * D_ + d4 * 4) = v;
  }
  __syncthreads();

  // stage 1: c1 = sigmoid(x @ enc_w1 + enc_b1)   K=128, N=256
  { v8f a = wmma_gemm_tile(sx, D_, p.enc_w1 + (size_t)e * D_ * H_, H_, wave * 16, D_);
    store_tile_lds(sh, H_, wave * 16, a, p.enc_b1 + e * H_, true); }
  __syncthreads();
  // stage 2: ctx = c1 @ enc_w2 + enc_b2          K=256, N=64
  if (wave < 4) {
    v8f a = wmma_gemm_tile(sh, H_, p.enc_w2 + (size_t)e * H_ * C_, C_, wave * 16, H_);
    store_tile_lds(sctx, C_, wave * 16, a, p.enc_b2 + e * C_, false);
  }
  __syncthreads();
  // stage 3: q = ctx @ qz0_w + qz0_b             K=64, N=128
  if (wave < 8) {
    v8f a = wmma_gemm_tile(sctx, C_, p.qz0_w + (size_t)e * C_ * 2 * L_, 2 * L_, wave * 16, C_);
    store_tile_lds(sq, 2 * L_, wave * 16, a, p.qz0_b + e * 2 * L_, false);
  }
  __syncthreads();

  // stage 4: z0 / z_in, build lat = [z_in | a | x], per-row KL
  for (int idx = tid; idx < 16 * L_; idx += 512) {
    const int row = idx >> 6, l = idx & (L_ - 1);
    const int b = r0 + row;
    const float qm = sq[row * 2 * L_ + l];
    const float ql = sq[row * 2 * L_ + L_ + l];
    float zin;
    if (t == 0) {
      const float n = (b < B_) ? p.z0_noise[(((size_t)t * E_ + e) * B_ + b) * L_ + l] : 0.f;
      zin = qm + __expf(ql) * n;
    } else {
      zin = (b < B_) ? p.zstate[((size_t)e * B_ + b) * L_ + l] : 0.f;
    }
    slat[row * 208 + l] = zin;
  }
  for (int i = tid; i < 16 * A_; i += 512) {
    const int row = i >> 4, c = i & (A_ - 1);
    const int b = r0 + row;
    slat[row * 208 + L_ + c] =
        (b < B_) ? p.actions[((size_t)e * S_ + (size_t)t * B_ + b) * A_ + c] : 0.f;
  }
  for (int i = tid; i < 16 * (D_ / 4); i += 512) {
    const int row = i >> 5, c4 = i & 31;
    *(v4f*)(slat + row * 208 + L_ + A_ + c4 * 4) = *(const v4f*)(sx + row * D_ + c4 * 4);
  }
  float klrow = 0.f;
  if (tid < 16) {
    for (int l = 0; l < L_; ++l) {
      const float qm = sq[tid * 2 * L_ + l];
      const float ql = sq[tid * 2 * L_ + L_ + l];
      const float m0 = p.pz0_mean[e * L_ + l];
      const float s0 = p.pz0_logstd[e * L_ + l];
      const float dm = qm - m0;
      klrow += s0 - ql + (__expf(2.f * ql) + dm * dm) / (2.f * __expf(2.f * s0)) - 0.5f;
    }
  }
  __syncthreads();

  // stage 5: z_enc = lat @ act_w + act_b         K=208, N=64
  if (wave < 4) {
    v8f a = wmma_gemm_tile(slat, 208, p.act_w + (size_t)e * 208 * L_, L_, wave * 16, 208);
    store_tile_lds(szc, L_, wave * 16, a, p.act_b + e * L_, false);
  }
  __syncthreads();
  for (int i = tid; i < 16 * (L_ / 4); i += 512)
    *(v4f*)(szh + i * 4) = *(const v4f*)(szc + i * 4);
  __syncthreads();

  // stage 6: SDE solve (midpoint-style with NSUB=2 substeps)
  float* fz  = sm + OFF_FZ0;
  float* gz  = sm + OFF_GZ0;
  float* fzn = sm + OFF_FZ1;
  float* gzn = sm + OFF_GZ1;
  float lf = drift_eval(p, sm, szc, fz, gz);
  float lc = 0.f, lh = 0.f;
  for (int k = 0; k < NSUB_; ++k) {
    for (int i = tid; i < 16 * (L_ / 4); i += 512) {
      const int row = i >> 4;
      const int b = r0 + row;
      v4f dw = {0.f, 0.f, 0.f, 0.f};
      if (b < B_)
        dw = *(const v4f*)(p.dW + ((((size_t)t * NSUB_ + k) * (E_ * B_)) +
                                   (size_t)e * B_ + b) * L_ + (i & 15) * 4) * SQDT_;
      const v4f zc4 = *(const v4f*)(szc + i * 4);
      const v4f zh4 = *(const v4f*)(szh + i * 4);
      const v4f fz4 = *(const v4f*)(fz + i * 4);
      const v4f gz4 = *(const v4f*)(gz + i * 4);
      *(v4f*)(sdw + i * 4) = dw;
      *(v4f*)(szh + i * 4) = 2.f * zc4 - zh4 + fz4 * DT_ + gz4 * dw;  // zh_n (in place)
    }
    if (tid < 16) lh = 2.f * lc - lh + lf * DT_;
    __syncthreads();
    const float lfn = drift_eval(p, sm, szh, fzn, gzn);
    for (int i = tid; i < 16 * (L_ / 4); i += 512) {
      const v4f fz4  = *(const v4f*)(fz + i * 4);
      const v4f fzn4 = *(const v4f*)(fzn + i * 4);
      const v4f gz4  = *(const v4f*)(gz + i * 4);
      const v4f gzn4 = *(const v4f*)(gzn + i * 4);
      const v4f dw4  = *(const v4f*)(sdw + i * 4);
      v4f zc4 = *(const v4f*)(szc + i * 4);
      zc4 += 0.5f * (fz4 + fzn4) * DT_ + 0.5f * (gz4 + gzn4) * dw4;
      *(v4f*)(szc + i * 4) = zc4;
    }
    if (tid < 16) lc += 0.5f * (lf + lfn) * DT_;
    __syncthreads();
    float* tmp = fz; fz = fzn; fzn = tmp;
    tmp = gz; gz = gzn; gzn = tmp;
    lf = lfn;
  }

  // stage 7: projection: L->H (sig) -> H (sig) -> D, store to output
  { v8f a = wmma_gemm_tile(szc, L_, p.pw1 + (size_t)e * L_ * H_, H_, wave * 16, L_);
    store_tile_lds(sh, H_, wave * 16, a, p.pb1 + e * H_, true); }
  __syncthreads();
  { v8f a = wmma_gemm_tile(sh, H_, p.pw2 + (size_t)e * H_ * H_, H_, wave * 16, H_);
    store_tile_lds(sh2, H_, wave * 16, a, p.pb2 + e * H_, true); }
  __syncthreads();
  if (wave < 8) {
    v8f a = wmma_gemm_tile(sh2, H_, p.pw3 + (size_t)e * H_ * D_, D_, wave * 16, H_);
    const int lane = tid & 31;
    const int mb   = (lane >> 4) << 3;
    const int col  = wave * 16 + (lane & 15);
    const float bv = p.pb3[e * D_ + col];
#pragma unroll
    for (int i = 0; i < 8; ++i) {
      const int b = r0 + mb + i;
      if (b < B_)
        p.out[E_ + ((size_t)e * S_ + (size_t)t * B_ + b) * D_ + col] = a[i] + bv;
    }
  }

  // stage 8: persist z state (float4) and accumulate logqp contributions
  for (int i = tid; i < 16 * (L_ / 4); i += 512) {
    const int row = i >> 4;
    const int b = r0 + row;
    if (b < B_)
      *(v4f*)(p.zstate + ((size_t)e * B_ + b) * L_ + (i & 15) * 4) =
          *(const v4f*)(szc + i * 4);
  }
  if (tid < 16) {
    const int b = r0 + tid;
    if (b < B_) {
      const float v = klrow + lc;
      if (t == 0) p.acc[e * B_ + b] = v;
      else        p.acc[e * B_ + b] += v;
    }
  }
}

__global__ void finalize_kernel(const float* __restrict__ acc, float* __restrict__ out) {
  const int e = threadIdx.x;
  if (e < E_) {
    float s = 0.f;
    for (int b = 0; b < B_; ++b) s += acc[e * B_ + b];
    out[e] = s / (float)B_;
  }
}

extern "C" void kernel_launch(void* const* d_in, const int* in_sizes, int n_in,
                              void* d_out, int out_size, void* d_ws, size_t ws_size,
                              hipStream_t stream) {
  (void)in_sizes; (void)n_in; (void)out_size; (void)ws_size;
  Params p;
  p.xs         = (const float*)d_in[0];
  p.actions    = (const float*)d_in[1];
  p.z0_noise   = (const float*)d_in[2];
  p.dW         = (const float*)d_in[3];
  p.enc_w1     = (const float*)d_in[4];
  p.enc_b1     = (const float*)d_in[5];
  p.enc_w2     = (const float*)d_in[6];
  p.enc_b2     = (const float*)d_in[7];
  p.qz0_w      = (const float*)d_in[8];
  p.qz0_b      = (const float*)d_in[9];
  p.act_w      = (const float*)d_in[10];
  p.act_b      = (const float*)d_in[11];
  p.fw1        = (const float*)d_in[12];
  p.fb1        = (const float*)d_in[13];
  p.fw2        = (const float*)d_in[14];
  p.fb2        = (const float*)d_in[15];
  p.fw3        = (const float*)d_in[16];
  p.fb3        = (const float*)d_in[17];
  p.hw1        = (const float*)d_in[18];
  p.hb1        = (const float*)d_in[19];
  p.hw2        = (const float*)d_in[20];
  p.hb2        = (const float*)d_in[21];
  p.hw3        = (const float*)d_in[22];
  p.hb3        = (const float*)d_in[23];
  p.gw1        = (const float*)d_in[24];
  p.gb1        = (const float*)d_in[25];
  p.gw2        = (const float*)d_in[26];
  p.gb2        = (const float*)d_in[27];
  p.pw1        = (const float*)d_in[28];
  p.pb1        = (const float*)d_in[29];
  p.pw2        = (const float*)d_in[30];
  p.pb2        = (const float*)d_in[31];
  p.pw3        = (const float*)d_in[32];
  p.pb3        = (const float*)d_in[33];
  p.pz0_mean   = (const float*)d_in[34];
  p.pz0_logstd = (const float*)d_in[35];
  p.out    = (float*)d_out;
  p.zstate = (float*)d_ws;
  p.acc    = (float*)d_ws + (size_t)E_ * B_ * L_;

  const size_t smem = (size_t)SMEM_FLOATS * sizeof(float);
  for (int t = 0; t < T_; ++t) {
    p.t = t;
    sde_step_kernel<<<dim3(E_ * 4), dim3(512), smem, stream>>>(p);
  }
  finalize_kernel<<<dim3(1), dim3(32), 0, stream>>>(p.acc, p.out);
}